// MultiScaleKeypointNet_33328946217682
// MI455X (gfx1250) — compile-verified
//
#include <hip/hip_runtime.h>
#include <hip/hip_bf16.h>

typedef __attribute__((ext_vector_type(2))) float v2f;
typedef __attribute__((ext_vector_type(8))) float v8f;

#define BN_EPS 1e-5f

static __device__ __forceinline__ int imin_(int a, int b) { return a < b ? a : b; }
static __device__ __forceinline__ int imax_(int a, int b) { return a > b ? a : b; }

// ---------------------------------------------------------------------------
// Direct 3x3 conv (pad=1), compile-time dims. Used for the small offset convs
// (Cout=18) where WMMA tiling is not worth it.
// ---------------------------------------------------------------------------
template <int Cin, int H, int W, int Ho, int Wo, int STRIDE>
__global__ void k_conv_direct(const float* __restrict__ in, const float* __restrict__ w,
                              const float* __restrict__ bias, float* __restrict__ out,
                              int B, int Cout) {
  int idx = blockIdx.x * blockDim.x + threadIdx.x;
  int total = B * Cout * Ho * Wo;
  if (idx >= total) return;
  int xo = idx % Wo;
  int t = idx / Wo;
  int yo = t % Ho;
  t /= Ho;
  int co = t % Cout;
  int n = t / Cout;
  float acc = bias[co];
  int oy = yo * STRIDE - 1;
  int ox = xo * STRIDE - 1;
  for (int ci = 0; ci < Cin; ++ci) {
    const float* ip = in + ((size_t)(n * Cin + ci)) * H * W;
    const float* wp = w + ((size_t)co * Cin + ci) * 9;
    __builtin_prefetch(ip + (oy + 3) * W + ox, 0, 0);  // next-row prefetch
#pragma unroll
    for (int ky = 0; ky < 3; ++ky) {
      int iy = oy + ky;
      if (iy < 0 || iy >= H) continue;
#pragma unroll
      for (int kx = 0; kx < 3; ++kx) {
        int ix = ox + kx;
        if (ix < 0 || ix >= W) continue;
        acc += ip[iy * W + ix] * wp[ky * 3 + kx];
      }
    }
  }
  out[idx] = acc;
}

// ---------------------------------------------------------------------------
// Fused: deform_conv(x, offset1, w_d1, stride=4) + conv1x1(avgpool4(x), w_skip)
// Cin=3 so direct VALU (27-deep GEMM is below WMMA utility). One thread per
// output spatial position, 64 cout accumulators in registers.
// ---------------------------------------------------------------------------
__global__ void k_deform1_skip(const float* __restrict__ x, const float* __restrict__ off,
                               const float* __restrict__ wd, const float* __restrict__ bd,
                               const float* __restrict__ wsk, const float* __restrict__ bsk,
                               float* __restrict__ out, int B) {
  const int H = 640, W = 640, Ho = 160, Wo = 160;
  int idx = blockIdx.x * blockDim.x + threadIdx.x;
  int total = B * Ho * Wo;
  if (idx >= total) return;
  int xo = idx % Wo;
  int yo = (idx / Wo) % Ho;
  int n = idx / (Wo * Ho);

  // avgpool4 of the 3 input channels at this output position
  float avg[3];
#pragma unroll
  for (int ci = 0; ci < 3; ++ci) {
    const float* pc = x + ((size_t)(n * 3 + ci)) * H * W;
    float s = 0.f;
#pragma unroll
    for (int r = 0; r < 4; ++r)
#pragma unroll
      for (int c = 0; c < 4; ++c)
        s += pc[(yo * 4 + r) * W + (xo * 4 + c)];
    avg[ci] = s * (1.f / 16.f);
  }

  float acc[64];
#pragma unroll
  for (int co = 0; co < 64; ++co)
    acc[co] = bd[co] + bsk[co] + wsk[co * 3 + 0] * avg[0] + wsk[co * 3 + 1] * avg[1] +
              wsk[co * 3 + 2] * avg[2];

  for (int tap = 0; tap < 9; ++tap) {
    float dy = off[((size_t)(n * 18 + 2 * tap) * Ho + yo) * Wo + xo];
    float dx = off[((size_t)(n * 18 + 2 * tap + 1) * Ho + yo) * Wo + xo];
    float ys = (float)(yo * 4 - 1 + tap / 3) + dy;
    float xs = (float)(xo * 4 - 1 + tap % 3) + dx;
    float y0f = floorf(ys), x0f = floorf(xs);
    float wy1 = ys - y0f, wx1 = xs - x0f;
    int iy0 = (int)y0f, ix0 = (int)x0f;
    int iy1 = iy0 + 1, ix1 = ix0 + 1;
    bool vy0 = (iy0 >= 0) && (iy0 < H), vy1 = (iy1 >= 0) && (iy1 < H);
    bool vx0 = (ix0 >= 0) && (ix0 < W), vx1 = (ix1 >= 0) && (ix1 < W);
    float w00 = (1.f - wy1) * (1.f - wx1) * ((vy0 && vx0) ? 1.f : 0.f);
    float w01 = (1.f - wy1) * wx1 * ((vy0 && vx1) ? 1.f : 0.f);
    float w10 = wy1 * (1.f - wx1) * ((vy1 && vx0) ? 1.f : 0.f);
    float w11 = wy1 * wx1 * ((vy1 && vx1) ? 1.f : 0.f);
    int cy0 = imin_(imax_(iy0, 0), H - 1), cy1 = imin_(imax_(iy1, 0), H - 1);
    int cx0 = imin_(imax_(ix0, 0), W - 1), cx1 = imin_(imax_(ix1, 0), W - 1);
    float s[3];
#pragma unroll
    for (int ci = 0; ci < 3; ++ci) {
      const float* pc = x + ((size_t)(n * 3 + ci)) * H * W;
      s[ci] = w00 * pc[cy0 * W + cx0] + w01 * pc[cy0 * W + cx1] + w10 * pc[cy1 * W + cx0] +
              w11 * pc[cy1 * W + cx1];
    }
#pragma unroll
    for (int co = 0; co < 64; ++co)
      acc[co] += wd[(co * 3 + 0) * 9 + tap] * s[0] + wd[(co * 3 + 1) * 9 + tap] * s[1] +
                 wd[(co * 3 + 2) * 9 + tap] * s[2];
  }
  // base + constant channel offsets -> immediate-offset stores
  float* ob = out + (((size_t)n * 64) * Ho + yo) * Wo + xo;
#pragma unroll
  for (int co = 0; co < 64; ++co) ob[co * (Ho * Wo)] = acc[co];
}

// ---------------------------------------------------------------------------
// Implicit-GEMM 3x3 conv, Cin=Cout=64, stride 1, pad 1, via
// V_WMMA_F32_16X16X4_F32.  Workgroup (256 thr = 8 waves) computes a
// 64-cout x 64-pixel tile of one output row.  Input patch staged in LDS,
// weights streamed as contiguous b64 loads (K ordered as ci*9+tap).
// Per wave-tile: M=16 couts, N=16 pixels, K=576 -> 144 WMMA issues.
//
// B-operand LDS addressing is periodic in K with period 36 (9 steps): the 18
// per-step addresses are hoisted (index div/mod done once), and the 16x9 K
// structure is fully unrolled so every ds_load_b32 / weight global_load_b64
// uses a compile-time immediate offset (group stride 3168 B / 144 B).
// ---------------------------------------------------------------------------
template <int H, int W>
__global__ void __launch_bounds__(256) k_conv3x3_wmma(const float* __restrict__ in,
                                                      const float* __restrict__ w,
                                                      float* __restrict__ out, int B) {
  const int C = 64;
  const int PW = 66;                // 64 pixels + 2 halo
  __shared__ float P[64 * 3 * PW];  // 50688 B
  const int nXT = (W + 63) / 64;
  int bx = blockIdx.x;
  int xt = bx % nXT;
  int y = (bx / nXT) % H;
  int n = bx / (nXT * H);
  int x0 = xt * 64;
  int tid = threadIdx.x;

  // stage input patch (zero-padded halo)
  for (int i = tid; i < 64 * 3 * PW; i += 256) {
    int ci = i / (3 * PW);
    int r = (i % (3 * PW)) / PW;
    int c = i % PW;
    int gy = y + r - 1;
    int gx = x0 + c - 1;
    float v = 0.f;
    if (gy >= 0 && gy < H && gx >= 0 && gx < W)
      v = in[((size_t)(n * C + ci) * H + gy) * W + gx];
    P[i] = v;
  }
  __syncthreads();

  int wave = tid >> 5;
  int lane = tid & 31;
  int m = lane & 15;   // A row (cout), B col (pixel), D col
  int hi = lane >> 4;  // K-half select
  int ct = wave & 3;
  int ptBase = (wave >> 2) * 2;
  int co = ct * 16 + m;
  const float* wco = w + (size_t)co * 576 + 2 * hi;  // K = ci*9+tap contiguous

  // Per-lane B addresses for one 36-K period (9 steps), hoisted.
  int addrX[9], addrY[9];
#pragma unroll
  for (int s = 0; s < 9; ++s) {
    int k0 = s * 4 + 2 * hi;
    int ci0 = k0 / 9, t0 = k0 - ci0 * 9, ky0 = t0 / 3, kx0 = t0 - ky0 * 3;
    int k1 = k0 + 1;
    int ci1 = k1 / 9, t1 = k1 - ci1 * 9, ky1 = t1 / 3, kx1 = t1 - ky1 * 3;
    addrX[s] = (ci0 * 3 + ky0) * PW + kx0;
    addrY[s] = (ci1 * 3 + ky1) * PW + kx1;
  }

  for (int pti = 0; pti < 2; ++pti) {
    int pt = ptBase + pti;
    int p = pt * 16 + m;  // pixel within 64-wide tile
    v8f acc = {0.f, 0.f, 0.f, 0.f, 0.f, 0.f, 0.f, 0.f};
#pragma unroll
    for (int grp = 0; grp < 16; ++grp) {  // 4 input channels per group
#pragma unroll
      for (int s = 0; s < 9; ++s) {
        v2f a = *(const v2f*)(wco + grp * 36 + s * 4);  // imm-offset b64
        v2f b;
        b.x = P[addrX[s] + p + grp * (4 * 3 * PW)];  // imm-offset ds_b32
        b.y = P[addrY[s] + p + grp * (4 * 3 * PW)];
        acc = __builtin_amdgcn_wmma_f32_16x16x4_f32(false, a, false, b, (short)0, acc,
                                                    false, false);
      }
    }
    int px = x0 + pt * 16 + m;
    if (px < W) {
      // base + constant per-channel offsets
      float* ob = out + ((size_t)(n * C + ct * 16 + 8 * hi) * H + y) * W + px;
#pragma unroll
      for (int v = 0; v < 8; ++v) ob[v * (H * W)] = acc[v];
    }
  }
}

// ---------------------------------------------------------------------------
// BatchNorm (affine=False, batch stats) helpers
// ---------------------------------------------------------------------------
__global__ void k_zero(float* p, int nElems) {
  int i = blockIdx.x * blockDim.x + threadIdx.x;
  if (i < nElems) p[i] = 0.f;
}

template <int C, int HW>
__global__ void k_bn_reduce(const float* __restrict__ x, float* __restrict__ stats) {
  int plane = blockIdx.x;  // (n, c) plane
  int c = plane % C;
  const float* px = x + (size_t)plane * HW;
  float s = 0.f, s2 = 0.f;
  for (int i = threadIdx.x; i < HW; i += 256) {
    float v = px[i];
    s += v;
    s2 += v * v;
  }
  __shared__ float sh[512];
  sh[threadIdx.x] = s;
  sh[256 + threadIdx.x] = s2;
  __syncthreads();
  for (int o = 128; o > 0; o >>= 1) {
    if (threadIdx.x < (unsigned)o) {
      sh[threadIdx.x] += sh[threadIdx.x + o];
      sh[256 + threadIdx.x] += sh[256 + threadIdx.x + o];
    }
    __syncthreads();
  }
  if (threadIdx.x == 0) {
    atomicAdd(&stats[c], sh[0]);
    atomicAdd(&stats[C + c], sh[256]);
  }
}

template <int C, int HW>
__global__ void k_bn_apply_relu(const float* __restrict__ x, const float* __restrict__ stats,
                                float* __restrict__ out, int B) {
  int i = blockIdx.x * blockDim.x + threadIdx.x;
  int total = B * C * HW;
  if (i >= total) return;
  int c = (i / HW) % C;
  float N = (float)B * (float)HW;
  float mean = stats[c] / N;
  float var = stats[C + c] / N - mean * mean;
  float r = rsqrtf(var + BN_EPS);
  float v = (x[i] - mean) * r;
  out[i] = v > 0.f ? v : 0.f;
}

// ---------------------------------------------------------------------------
// Deformable conv, Cin=64, 3x3, pad=1, compile-time stride/Cout/dims, via WMMA.
// Workgroup (256 thr) handles 16 output pixels of one row x all Cout:
//  1) 144 threads bilinear-sample the 9x64x16 im2col patch into LDS
//     (layout S[k=ci*9+tap][pixel], 36 KB; addressing linear in K)
//  2) each wave runs 16cout x 16pixel WMMA GEMM tiles (K=576, fully unrolled
//     so ds/global loads use immediate offsets), bias fused.
// ---------------------------------------------------------------------------
template <int Cout, int Hin, int Win, int Ho, int Wo, int STRIDE>
__global__ void __launch_bounds__(256) k_deform_wmma(
    const float* __restrict__ in, const float* __restrict__ off, const float* __restrict__ w,
    const float* __restrict__ bias, float* __restrict__ out, int B) {
  const int Cin = 64;
  __shared__ float S[576 * 16];  // 36864 B
  const int nXT = (Wo + 15) / 16;
  int bx = blockIdx.x;
  int xt = bx % nXT;
  int yo = (bx / nXT) % Ho;
  int n = bx / (nXT * Ho);
  int xo0 = xt * 16;
  int tid = threadIdx.x;

  if (tid < 144) {  // one thread per (tap, pixel)
    int tap = tid / 16;
    int p = tid % 16;
    int xo = xo0 + p;
    if (xo >= Wo) {
      for (int ci = 0; ci < Cin; ++ci) S[(ci * 9 + tap) * 16 + p] = 0.f;
    } else {
      float dy = off[((size_t)(n * 18 + 2 * tap) * Ho + yo) * Wo + xo];
      float dx = off[((size_t)(n * 18 + 2 * tap + 1) * Ho + yo) * Wo + xo];
      float ys = (float)(yo * STRIDE - 1 + tap / 3) + dy;
      float xs = (float)(xo * STRIDE - 1 + tap % 3) + dx;
      float y0f = floorf(ys), x0f = floorf(xs);
      float wy1 = ys - y0f, wx1 = xs - x0f;
      int iy0 = (int)y0f, ix0 = (int)x0f;
      int iy1 = iy0 + 1, ix1 = ix0 + 1;
      bool vy0 = (iy0 >= 0) && (iy0 < Hin), vy1 = (iy1 >= 0) && (iy1 < Hin);
      bool vx0 = (ix0 >= 0) && (ix0 < Win), vx1 = (ix1 >= 0) && (ix1 < Win);
      float w00 = (1.f - wy1) * (1.f - wx1) * ((vy0 && vx0) ? 1.f : 0.f);
      float w01 = (1.f - wy1) * wx1 * ((vy0 && vx1) ? 1.f : 0.f);
      float w10 = wy1 * (1.f - wx1) * ((vy1 && vx0) ? 1.f : 0.f);
      float w11 = wy1 * wx1 * ((vy1 && vx1) ? 1.f : 0.f);
      int cy0 = imin_(imax_(iy0, 0), Hin - 1), cy1 = imin_(imax_(iy1, 0), Hin - 1);
      int cx0 = imin_(imax_(ix0, 0), Win - 1), cx1 = imin_(imax_(ix1, 0), Win - 1);
      int o00 = cy0 * Win + cx0, o01 = cy0 * Win + cx1;
      int o10 = cy1 * Win + cx0, o11 = cy1 * Win + cx1;
      const float* base = in + (size_t)n * Cin * Hin * Win;
      for (int ci = 0; ci < Cin; ++ci) {
        const float* pc = base + (size_t)ci * Hin * Win;
        S[(ci * 9 + tap) * 16 + p] =
            w00 * pc[o00] + w01 * pc[o01] + w10 * pc[o10] + w11 * pc[o11];
      }
    }
  }
  __syncthreads();

  int wave = tid >> 5;
  int lane = tid & 31;
  int m = lane & 15;
  int hi = lane >> 4;
  const float* sb = S + 2 * hi * 16 + m;  // B base; step offsets are linear
  const int nCt = Cout >> 4;
  for (int ct = wave; ct < nCt; ct += 8) {
    int co = ct * 16 + m;
    const float* wco = w + (size_t)co * 576 + 2 * hi;  // (ci,tap) contiguous = K
    v8f acc = {0.f, 0.f, 0.f, 0.f, 0.f, 0.f, 0.f, 0.f};
#pragma unroll
    for (int step = 0; step < 144; ++step) {
      v2f a = *(const v2f*)(wco + step * 4);  // imm-offset b64
      v2f b;
      b.x = sb[step * 64];       // imm-offset ds_b32 (step*256 B)
      b.y = sb[step * 64 + 16];  // imm-offset ds_b32
      acc = __builtin_amdgcn_wmma_f32_16x16x4_f32(false, a, false, b, (short)0, acc, false,
                                                  false);
    }
    int xo = xo0 + m;
    if (xo < Wo) {
      float* ob = out + (((size_t)n * Cout + ct * 16 + 8 * hi) * Ho + yo) * Wo + xo;
      const float* bb = bias + ct * 16 + 8 * hi;
#pragma unroll
      for (int v = 0; v < 8; ++v) ob[v * (Ho * Wo)] = acc[v] + bb[v];
    }
  }
}

// ---------------------------------------------------------------------------
// Host-side orchestration
// ---------------------------------------------------------------------------
extern "C" void kernel_launch(void* const* d_in, const int* in_sizes, int n_in, void* d_out,
                              int out_size, void* d_ws, size_t ws_size, hipStream_t stream) {
  (void)in_sizes;
  (void)n_in;
  (void)out_size;
  (void)ws_size;
  const float* x = (const float*)d_in[0];
  const float* w_off1 = (const float*)d_in[1];
  const float* b_off1 = (const float*)d_in[2];
  const float* w_d1 = (const float*)d_in[3];
  const float* b_d1 = (const float*)d_in[4];
  const float* w_skip = (const float*)d_in[5];
  const float* b_skip = (const float*)d_in[6];
  const float* w_bl1 = (const float*)d_in[7];
  const float* w_bl2 = (const float*)d_in[8];
  const float* w_off2 = (const float*)d_in[9];
  const float* b_off2 = (const float*)d_in[10];
  const float* w_d2 = (const float*)d_in[11];
  const float* b_d2 = (const float*)d_in[12];
  const float* w_off3 = (const float*)d_in[13];
  const float* b_off3 = (const float*)d_in[14];
  const float* w_d3 = (const float*)d_in[15];
  const float* b_d3 = (const float*)d_in[16];

  const int B = 8;
  const size_t PLANE14 = (size_t)B * 64 * 160 * 160;  // 13107200
  float* ws = (float*)d_ws;
  float* bufA = ws;                           // t1, then f
  float* bufB = ws + PLANE14;                 // raw conv outputs
  float* bufC = ws + 2 * PLANE14;             // offset maps
  float* stats = ws + 2 * PLANE14 + 3686400;  // 128 floats: sum | sumsq

  float* feat14 = (float*)d_out;
  float* feat18 = feat14 + PLANE14;
  float* feat116 = feat18 + (size_t)B * 64 * 80 * 80;

  // 1) offset1 = conv(x, w_off1, stride=4)            [8,18,160,160]
  {
    int total = B * 18 * 160 * 160;
    k_conv_direct<3, 640, 640, 160, 160, 4>
        <<<(total + 255) / 256, 256, 0, stream>>>(x, w_off1, b_off1, bufC, B, 18);
  }
  // 2) t1 = deform_conv(x, offset1) + skip            [8,64,160,160]
  {
    int total = B * 160 * 160;
    k_deform1_skip<<<(total + 255) / 256, 256, 0, stream>>>(x, bufC, w_d1, b_d1, w_skip,
                                                            b_skip, bufA, B);
  }
  // 3) conv bl1 (WMMA) -> bufB
  k_conv3x3_wmma<160, 160><<<B * 160 * 3, 256, 0, stream>>>(bufA, w_bl1, bufB, B);
  // 4) bn_relu -> f (bufA)
  k_zero<<<1, 128, 0, stream>>>(stats, 128);
  k_bn_reduce<64, 25600><<<B * 64, 256, 0, stream>>>(bufB, stats);
  k_bn_apply_relu<64, 25600>
      <<<(B * 64 * 25600 + 255) / 256, 256, 0, stream>>>(bufB, stats, bufA, B);
  // 5) conv bl2 (WMMA) -> bufB
  k_conv3x3_wmma<160, 160><<<B * 160 * 3, 256, 0, stream>>>(bufA, w_bl2, bufB, B);
  // 6) bn_relu -> feat_1_4 (d_out)
  k_zero<<<1, 128, 0, stream>>>(stats, 128);
  k_bn_reduce<64, 25600><<<B * 64, 256, 0, stream>>>(bufB, stats);
  k_bn_apply_relu<64, 25600>
      <<<(B * 64 * 25600 + 255) / 256, 256, 0, stream>>>(bufB, stats, feat14, B);
  // 7) offset2 = conv(feat_1_4, stride=2)             [8,18,80,80]
  {
    int total = B * 18 * 80 * 80;
    k_conv_direct<64, 160, 160, 80, 80, 2>
        <<<(total + 255) / 256, 256, 0, stream>>>(feat14, w_off2, b_off2, bufC, B, 18);
  }
  // 8) feat_1_8 = deform_conv(feat_1_4, offset2) (WMMA)
  k_deform_wmma<64, 160, 160, 80, 80, 2>
      <<<B * 80 * 5, 256, 0, stream>>>(feat14, bufC, w_d2, b_d2, feat18, B);
  // 9) offset3 = conv(feat_1_8, stride=2)             [8,18,40,40]
  {
    int total = B * 18 * 40 * 40;
    k_conv_direct<64, 80, 80, 40, 40, 2>
        <<<(total + 255) / 256, 256, 0, stream>>>(feat18, w_off3, b_off3, bufC, B, 18);
  }
  // 10) feat_1_16 = deform_conv(feat_1_8, offset3) (WMMA, Cout=256)
  k_deform_wmma<256, 80, 80, 40, 40, 2>
      <<<B * 40 * 3, 256, 0, stream>>>(feat18, bufC, w_d3, b_d3, feat116, B);
}